// LovaszSoftmaxLoss_39058432590541
// MI455X (gfx1250) — compile-verified
//
#include <hip/hip_runtime.h>
#include <hip/hip_bf16.h>
#include <stdint.h>

typedef unsigned int u32;
typedef __attribute__((ext_vector_type(2))) float v2f;
typedef __attribute__((ext_vector_type(8))) float v8f;

#define NPIX (512 * 512)   // 262144 pixels per (b) image
#define NB 8
#define NC 21
#define NROWS (NB * NC)    // 168 independent rows to sort

// ---------------------------------------------------------------------------
// Zero the scalar output (harness poisons d_out; we must be deterministic).
__global__ void lovasz_init(float* out) { out[0] = 0.0f; }

// ---------------------------------------------------------------------------
// Per-row key/payload builder: softmax over 21 classes (whole input lives in
// the 192MB L2, so the x21 per-row recompute is nearly free), error ->
// 30-bit descending-order radix key, label -> payload.
__global__ __launch_bounds__(256)
void lovasz_build_keys(const float* __restrict__ logits,
                       const int* __restrict__ target,
                       u32* __restrict__ keys, u32* __restrict__ pay,
                       int row0) {
  int row_local = blockIdx.y;
  int row = row0 + row_local;
  int b = row / NC, c = row % NC;
  int p = blockIdx.x * 256 + threadIdx.x;
  size_t bbase = (size_t)b * NC * NPIX;

  float v[NC];
#pragma unroll
  for (int j = 0; j < NC; ++j) v[j] = logits[bbase + (size_t)j * NPIX + p];
  float m = v[0];
#pragma unroll
  for (int j = 1; j < NC; ++j) m = fmaxf(m, v[j]);
  float s = 0.f, ec = 0.f;
#pragma unroll
  for (int j = 0; j < NC; ++j) {
    float e = __expf(v[j] - m);
    s += e;
    if (j == c) ec = e;
  }
  float prob = ec / s;
  int label = target[(size_t)b * NPIX + p];
  float fg = (label == c) ? 1.f : 0.f;
  float err = fabsf(fg - prob);                    // in [0,1] -> bits <= 0x3F800000
  u32 key = 0x3F800000u - __float_as_uint(err);    // ascending key == descending err
  size_t o = (size_t)row_local * NPIX + p;
  keys[o] = key;
  pay[o] = (u32)label;
}

// ---------------------------------------------------------------------------
// One wave (32 lanes) per row: stable LSD radix sort, 4x8-bit passes over the
// 30-bit key. LDS histogram (ds_add), shfl scan of 256 bins, and a
// ballot-grouped stable scatter (wave-ordered LDS bin counters keep pass
// stability, which reproduces jnp.argsort tie order exactly).
__global__ __launch_bounds__(32)
void lovasz_sort(u32* __restrict__ keysA, u32* __restrict__ payA,
                 u32* __restrict__ keysB, u32* __restrict__ payB) {
  int row = blockIdx.x;
  int lane = threadIdx.x;
  u32* ka = keysA + (size_t)row * NPIX;
  u32* pa = payA + (size_t)row * NPIX;
  u32* kb = keysB + (size_t)row * NPIX;
  u32* pb = payB + (size_t)row * NPIX;
  __shared__ u32 hist[256];
  __shared__ u32 offs[256];

  for (int pass = 0; pass < 4; ++pass) {
    int sh = pass * 8;
    u32* sk = (pass & 1) ? kb : ka;
    u32* sp = (pass & 1) ? pb : pa;
    u32* dk = (pass & 1) ? ka : kb;
    u32* dp = (pass & 1) ? pa : pb;

#pragma unroll
    for (int i = 0; i < 8; ++i) hist[lane * 8 + i] = 0;
    __syncthreads();

    for (int i = lane; i < NPIX; i += 32) {
      if (i + 2048 < NPIX) __builtin_prefetch(&sk[i + 2048], 0, 1);
      u32 d = (sk[i] >> sh) & 255u;
      atomicAdd(&hist[d], 1u);
    }
    __syncthreads();

    // exclusive scan of 256 bins: 8 bins/lane local + shfl scan across lanes
    u32 loc[8];
    u32 s = 0;
#pragma unroll
    for (int i = 0; i < 8; ++i) {
      loc[i] = s;
      s += hist[lane * 8 + i];
    }
    u32 inc = s;
#pragma unroll
    for (int d = 1; d < 32; d <<= 1) {
      u32 n = __shfl_up(inc, d, 32);
      if (lane >= d) inc += n;
    }
    u32 excl = inc - s;
#pragma unroll
    for (int i = 0; i < 8; ++i) offs[lane * 8 + i] = excl + loc[i];
    __syncthreads();

    u32 lt = (lane == 0) ? 0u : (0xFFFFFFFFu >> (32 - lane));
    for (int g = 0; g < NPIX; g += 32) {
      int i = g + lane;
      if (i + 2048 < NPIX) __builtin_prefetch(&sk[i + 2048], 0, 1);
      u32 k = sk[i], pld = sp[i];
      u32 d = (k >> sh) & 255u;
      // group lanes with the same digit via per-bit ballots
      u32 mask = 0xFFFFFFFFu;
#pragma unroll
      for (int bit = 0; bit < 8; ++bit) {
        u32 bal = (u32)__ballot((d >> bit) & 1);
        mask &= ((d >> bit) & 1) ? bal : ~bal;
      }
      u32 rank = __popc(mask & lt);
      u32 base = offs[d];  // wave-ordered LDS read precedes leader update
      u32 pos = base + rank;
      dk[pos] = k;
      dp[pos] = pld;
      if ((mask & lt) == 0)  // lowest lane of the digit group
        atomicAdd(&offs[d], (u32)__popc(mask));
    }
    __syncthreads();
  }
}

// ---------------------------------------------------------------------------
// One wave per row: Lovasz gradient + loss in a single streaming pass.
// 256-element inclusive prefix-sum chunks computed on the matrix core:
//   S = L (16x16 lower-triangular ones) @ X  via 4 chained
//   V_WMMA_F32_16X16X4_F32 K-slices; cross-column offsets via 16-wide shfl
//   scan. Loss uses  sum_k jacc[k]*(e[k]-e[k+1]),  e[N]:=0.
__global__ __launch_bounds__(32)
void lovasz_loss(const u32* __restrict__ keysA, const u32* __restrict__ payA,
                 float* __restrict__ out) {
  int row = blockIdx.x;
  int lane = threadIdx.x;
  const u32* keys = keysA + (size_t)row * NPIX;
  const u32* pay = payA + (size_t)row * NPIX;
  __shared__ float lds_t[256];
  __shared__ float lds_e[257];

  // tsum: sum of labels over the row (integer-valued, < 2^24 -> exact)
  float ts = 0.f;
  for (int i = lane; i < NPIX; i += 32) ts += (float)pay[i];
#pragma unroll
  for (int m = 16; m >= 1; m >>= 1) ts += __shfl_xor(ts, m, 32);

  int j = lane & 15;        // C/D column, A row, B column for this lane
  int half = lane >> 4;     // lane half selects K / M sub-ranges
  float base = 0.f, loss = 0.f;

  for (int chunk = 0; chunk < NPIX; chunk += 256) {
    // stage chunk into LDS (coalesced), element order = j*16 + i
#pragma unroll
    for (int u = 0; u < 8; ++u) {
      int idx = lane * 8 + u;
      u32 k = keys[chunk + idx];
      lds_e[idx] = __uint_as_float(0x3F800000u - k);
      lds_t[idx] = (float)pay[chunk + idx];
    }
    if (lane == 0) {
      lds_e[256] = (chunk + 256 < NPIX)
                       ? __uint_as_float(0x3F800000u - keys[chunk + 256])
                       : 0.f;
    }
    __syncthreads();

    // column-wise inclusive scan of 16 columns x 16 rows via WMMA
    v8f acc = {0.f, 0.f, 0.f, 0.f, 0.f, 0.f, 0.f, 0.f};
#pragma unroll
    for (int k0 = 0; k0 < 16; k0 += 4) {
      int kbse = k0 + 2 * half;  // A layout: K = vgpr + 2*(lane/16)
      v2f a, bv;
      a.x = (kbse <= j) ? 1.f : 0.f;      // L[i][k] = (k <= i)
      a.y = (kbse + 1 <= j) ? 1.f : 0.f;
      bv.x = lds_t[j * 16 + kbse];        // B[k][j] = X[k][j]
      bv.y = lds_t[j * 16 + kbse + 1];
      acc = __builtin_amdgcn_wmma_f32_16x16x4_f32(false, a, false, bv,
                                                  (short)0, acc, false, false);
    }

    // cross-column exclusive offsets from column sums S[15][:]
    float cs = __shfl(acc[7], 16 + j, 32);  // S[15][j] lives in lane 16+j, vgpr7
    float incj = cs;
#pragma unroll
    for (int d = 1; d < 16; d <<= 1) {
      float n = __shfl_up(incj, d, 16);
      if (j >= d) incj += n;
    }
    float exclj = incj - cs;
    float ctotal = __shfl(incj, 15, 16);

#pragma unroll
    for (int r = 0; r < 8; ++r) {
      int i = r + 8 * half;        // C/D layout: M = vgpr + 8*(lane/16)
      int off = j * 16 + i;        // position within chunk
      float P = acc[r] + exclj + base;          // inclusive cumsum of labels
      float kp1 = (float)(chunk + off + 1);
      float I = ts - P;                         // intersection
      float U = ts + kp1 - P;                   // union (always > 0)
      float jacc = 1.f - I / U;
      loss += jacc * (lds_e[off] - lds_e[off + 1]);
    }
    base += ctotal;
    __syncthreads();
  }

#pragma unroll
  for (int m = 16; m >= 1; m >>= 1) loss += __shfl_xor(loss, m, 32);
  if (lane == 0) atomicAdd(out, loss * (1.0f / (float)NROWS));
}

// ---------------------------------------------------------------------------
extern "C" void kernel_launch(void* const* d_in, const int* in_sizes, int n_in,
                              void* d_out, int out_size, void* d_ws,
                              size_t ws_size, hipStream_t stream) {
  const float* logits = (const float*)d_in[0];
  const int* target = (const int*)d_in[1];
  float* out = (float*)d_out;

  // ping+pong keys + ping+pong payload per in-flight row
  size_t per_row = (size_t)NPIX * 4u * 4u;  // 4 MB
  int R = (int)(ws_size / per_row);
  if (R < 1) R = 1;
  if (R > NROWS) R = NROWS;

  u32* keysA = (u32*)d_ws;
  u32* payA = keysA + (size_t)R * NPIX;
  u32* keysB = payA + (size_t)R * NPIX;
  u32* payB = keysB + (size_t)R * NPIX;

  lovasz_init<<<dim3(1), dim3(1), 0, stream>>>(out);

  for (int row0 = 0; row0 < NROWS; row0 += R) {
    int Rb = (NROWS - row0 < R) ? (NROWS - row0) : R;
    lovasz_build_keys<<<dim3(NPIX / 256, Rb), dim3(256), 0, stream>>>(
        logits, target, keysA, payA, row0);
    lovasz_sort<<<dim3(Rb), dim3(32), 0, stream>>>(keysA, payA, keysB, payB);
    // 4 passes: A->B->A->B->A, data ends in A
    lovasz_loss<<<dim3(Rb), dim3(32), 0, stream>>>(keysA, payA, out);
  }
}